// Net_7825430413945
// MI455X (gfx1250) — compile-verified
//
#include <hip/hip_runtime.h>
#include <math.h>

typedef __attribute__((ext_vector_type(2))) float v2f;
typedef __attribute__((ext_vector_type(8))) float v8f;

#define N_FEAT   50
#define HIDDEN   16
#define N_CLS    10

// ---------------- utility ----------------
__global__ void zero_fill(float* __restrict__ p, long n) {
  long i = (long)blockIdx.x * blockDim.x + threadIdx.x;
  if (i < n) p[i] = 0.0f;
}

// ---------------- graph norm ----------------
__global__ void deg_kernel(const int* __restrict__ dst, float* __restrict__ deg, int E) {
  int e = blockIdx.x * blockDim.x + threadIdx.x;
  if (e < E) atomicAdd(&deg[dst[e]], 1.0f);
}

__global__ void dinv_kernel(const float* __restrict__ deg, float* __restrict__ dinv, int n) {
  int i = blockIdx.x * blockDim.x + threadIdx.x;
  if (i < n) {
    float d = deg[i];
    dinv[i] = (d > 0.0f) ? rsqrtf(d) : 0.0f;
  }
}

__global__ void norm_kernel(const int* __restrict__ src, const int* __restrict__ dst,
                            const float* __restrict__ dinv, float* __restrict__ nrm, int E) {
  int e = blockIdx.x * blockDim.x + threadIdx.x;
  if (e < E) nrm[e] = dinv[src[e]] * dinv[dst[e]];
}

// ---------------- edge scatter-adds ----------------
// layer 1: 2 waves per edge (64 threads) cover 50 features.
// Edge id is wave-uniform -> readfirstlane makes src/dst/norm scalar loads.
__global__ void scatter50(const int* __restrict__ src, const int* __restrict__ dst,
                          const float* __restrict__ nrm, const float* __restrict__ x,
                          float* __restrict__ agg, int E) {
  long gt   = (long)blockIdx.x * blockDim.x + threadIdx.x;
  int  w    = (int)(gt >> 5);                 // global wave id (wave-uniform)
  int  lane = threadIdx.x & 31;
  int  e    = __builtin_amdgcn_readfirstlane(w >> 1);   // uniform edge id -> SGPR
  int  f    = ((w & 1) << 5) + lane;                    // 0..63
  if (e >= E) return;                                    // uniform exit
  if (f < N_FEAT) {
    int   s = src[e];                                    // scalar loads
    int   d = dst[e];
    float g = nrm[e];
    float v = x[(long)s * N_FEAT + f] * g;
    atomicAdd(&agg[(long)d * N_FEAT + f], v);
  }
}

// layer 2: 16 threads per edge cover 16 hidden features
__global__ void scatter16(const int* __restrict__ src, const int* __restrict__ dst,
                          const float* __restrict__ nrm, const float* __restrict__ h,
                          float* __restrict__ agg, int E) {
  long t = (long)blockIdx.x * blockDim.x + threadIdx.x;
  int e = (int)(t >> 4);
  int f = (int)(t & 15);
  if (e >= E) return;
  int s = src[e], d = dst[e];
  float v = h[(long)s * HIDDEN + f] * nrm[e];
  atomicAdd(&agg[(long)d * HIDDEN + f], v);
}

// ---------------- dense layer 1: h = relu(x@W10 + agg@W11 + b1) ----------------
// one wave per 16-node tile; N=16(=HIDDEN); K=50 padded to 52 (branch-free tail).
__global__ __launch_bounds__(256) void tag1_dense(
    const float* __restrict__ X, const float* __restrict__ AGG,
    const float* __restrict__ W0, const float* __restrict__ W1,
    const float* __restrict__ B, float* __restrict__ H, int n) {
  int wave = (int)((blockIdx.x * blockDim.x + threadIdx.x) >> 5);
  int lane = threadIdx.x & 31;
  int base = wave * 16;
  if (base >= n) return;                 // wave-uniform: EXEC stays all-ones

  int row   = lane & 15;                 // A: M row / B: N col
  int khalf = (lane >> 4) * 2;           // lanes 16-31 hold K+2,K+3
  int node  = base + row;
  node = node < n ? node : n - 1;        // clamp (no-op: n % 16 == 0)
  long xrow = (long)node * N_FEAT;
  bool lo   = (khalf == 0);

  v8f c = {};
  // ---- X @ W0 ----
#pragma unroll
  for (int k = 0; k < 48; k += 4) {
    v2f a = *(const v2f*)(X + xrow + k + khalf);          // aligned b64
    v2f b;
    b.x = W0[(k + khalf)     * HIDDEN + row];
    b.y = W0[(k + khalf + 1) * HIDDEN + row];
    c = __builtin_amdgcn_wmma_f32_16x16x4_f32(false, a, false, b, (short)0, c, false, false);
  }
  { // tail K = 48..51: logical K=50,51 are zero (khalf==2 lanes)
    v2f a0 = *(const v2f*)(X + xrow + 48);
    float b0x = W0[48 * HIDDEN + row];
    float b0y = W0[49 * HIDDEN + row];
    v2f a, b;
    a.x = lo ? a0.x : 0.0f;  a.y = lo ? a0.y : 0.0f;
    b.x = lo ? b0x  : 0.0f;  b.y = lo ? b0y  : 0.0f;
    c = __builtin_amdgcn_wmma_f32_16x16x4_f32(false, a, false, b, (short)0, c, false, false);
  }
  // ---- AGG @ W1 ----
#pragma unroll
  for (int k = 0; k < 48; k += 4) {
    v2f a = *(const v2f*)(AGG + xrow + k + khalf);
    v2f b;
    b.x = W1[(k + khalf)     * HIDDEN + row];
    b.y = W1[(k + khalf + 1) * HIDDEN + row];
    c = __builtin_amdgcn_wmma_f32_16x16x4_f32(false, a, false, b, (short)0, c, false, false);
  }
  {
    v2f a0 = *(const v2f*)(AGG + xrow + 48);
    float b0x = W1[48 * HIDDEN + row];
    float b0y = W1[49 * HIDDEN + row];
    v2f a, b;
    a.x = lo ? a0.x : 0.0f;  a.y = lo ? a0.y : 0.0f;
    b.x = lo ? b0x  : 0.0f;  b.y = lo ? b0y  : 0.0f;
    c = __builtin_amdgcn_wmma_f32_16x16x4_f32(false, a, false, b, (short)0, c, false, false);
  }

  int col = lane & 15;
  int mhi = (lane >> 4) * 8;
  float bias = B[col];
#pragma unroll
  for (int j = 0; j < 8; ++j) {
    int m = base + mhi + j;
    if (m < n) {
      float v = c[j] + bias;
      H[(long)m * HIDDEN + col] = v > 0.0f ? v : 0.0f;
    }
  }
}

// ---------------- dense layer 2 + log_softmax ----------------
// out = h@W20 + agg2@W21 + b2 ; N padded 10 -> 16; K = 16.
__global__ __launch_bounds__(256) void tag2_dense(
    const float* __restrict__ H, const float* __restrict__ AGG,
    const float* __restrict__ W0, const float* __restrict__ W1,
    const float* __restrict__ B, float* __restrict__ OUT, int n) {
  int wave = (int)((blockIdx.x * blockDim.x + threadIdx.x) >> 5);
  int lane = threadIdx.x & 31;
  int base = wave * 16;
  if (base >= n) return;

  int row   = lane & 15;
  int khalf = (lane >> 4) * 2;
  int node  = base + row;
  node = node < n ? node : n - 1;
  long hrow = (long)node * HIDDEN;
  bool bok  = row < N_CLS;
  int  wrow = bok ? row : (N_CLS - 1);   // clamp weight column, mask later

  v8f c = {};
#pragma unroll
  for (int k = 0; k < HIDDEN; k += 4) {
    int ka = k + khalf;
    v2f a = *(const v2f*)(H + hrow + ka);
    float b0 = W0[(ka)     * N_CLS + wrow];
    float b1 = W0[(ka + 1) * N_CLS + wrow];
    v2f b;
    b.x = bok ? b0 : 0.0f;  b.y = bok ? b1 : 0.0f;
    c = __builtin_amdgcn_wmma_f32_16x16x4_f32(false, a, false, b, (short)0, c, false, false);
  }
#pragma unroll
  for (int k = 0; k < HIDDEN; k += 4) {
    int ka = k + khalf;
    v2f a = *(const v2f*)(AGG + hrow + ka);
    float b0 = W1[(ka)     * N_CLS + wrow];
    float b1 = W1[(ka + 1) * N_CLS + wrow];
    v2f b;
    b.x = bok ? b0 : 0.0f;  b.y = bok ? b1 : 0.0f;
    c = __builtin_amdgcn_wmma_f32_16x16x4_f32(false, a, false, b, (short)0, c, false, false);
  }

  int col  = lane & 15;
  int mhi  = (lane >> 4) * 8;
  bool cok = col < N_CLS;
  float bias = cok ? B[col] : 0.0f;

#pragma unroll
  for (int j = 0; j < 8; ++j) {
    int m = base + mhi + j;
    float v = c[j] + bias;
    // row-wise log-softmax across each 16-lane group (cols 0..9 valid)
    float mx = cok ? v : -INFINITY;
    for (int off = 8; off >= 1; off >>= 1)
      mx = fmaxf(mx, __shfl_xor(mx, off, 16));
    float ex = cok ? expf(v - mx) : 0.0f;
    float sm = ex;
    for (int off = 8; off >= 1; off >>= 1)
      sm += __shfl_xor(sm, off, 16);
    float r = v - mx - logf(sm);
    if (cok && m < n) OUT[(long)m * N_CLS + col] = r;
  }
}

// ---------------- host launch ----------------
extern "C" void kernel_launch(void* const* d_in, const int* in_sizes, int n_in,
                              void* d_out, int out_size, void* d_ws, size_t ws_size,
                              hipStream_t stream) {
  const float* x    = (const float*)d_in[0];
  const int*   ei   = (const int*)d_in[1];
  const float* w1_0 = (const float*)d_in[2];
  const float* w1_1 = (const float*)d_in[3];
  const float* b1   = (const float*)d_in[4];
  const float* w2_0 = (const float*)d_in[5];
  const float* w2_1 = (const float*)d_in[6];
  const float* b2   = (const float*)d_in[7];
  float* out = (float*)d_out;

  int n = in_sizes[0] / N_FEAT;   // 100000
  int E = in_sizes[1] / 2;        // 1600000
  const int* src = ei;
  const int* dst = ei + E;

  // workspace layout (floats); zero-init region [deg|agg1|agg2] is contiguous
  float* ws   = (float*)d_ws;
  float* deg  = ws;
  float* agg1 = deg  + n;
  float* agg2 = agg1 + (long)n * N_FEAT;
  float* dinv = agg2 + (long)n * HIDDEN;
  float* nrm  = dinv + n;
  float* h    = nrm  + E;

  long zn = (long)n + (long)n * N_FEAT + (long)n * HIDDEN;
  zero_fill<<<(unsigned)((zn + 255) / 256), 256, 0, stream>>>(deg, zn);

  deg_kernel <<<(E + 255) / 256, 256, 0, stream>>>(dst, deg, E);
  dinv_kernel<<<(n + 255) / 256, 256, 0, stream>>>(deg, dinv, n);
  norm_kernel<<<(E + 255) / 256, 256, 0, stream>>>(src, dst, dinv, nrm, E);

  long t50 = (long)E * 64;
  scatter50<<<(unsigned)((t50 + 255) / 256), 256, 0, stream>>>(src, dst, nrm, x, agg1, E);

  int ntiles = (n + 15) / 16;
  int nblk   = (ntiles + 7) / 8;   // 8 waves (256 threads) per block
  tag1_dense<<<nblk, 256, 0, stream>>>(x, agg1, w1_0, w1_1, b1, h, n);

  long t16 = (long)E * 16;
  scatter16<<<(unsigned)((t16 + 255) / 256), 256, 0, stream>>>(src, dst, nrm, h, agg2, E);

  tag2_dense<<<nblk, 256, 0, stream>>>(h, agg2, w2_0, w2_1, b2, out, n);
}